// Repro_11879879544022
// MI455X (gfx1250) — compile-verified
//
#include <hip/hip_runtime.h>
#include <math.h>

typedef __attribute__((ext_vector_type(2))) float v2f;
typedef __attribute__((ext_vector_type(4))) float v4f;
typedef __attribute__((ext_vector_type(8))) float v8f;

#define LEAKY_SLOPE 0.2f

template <int PER> struct VecT;
template <> struct VecT<4> { typedef v4f T; };
template <> struct VecT<2> { typedef v2f T; };

__device__ __forceinline__ float gelu_erf(float x) {
  // exact (erf) GELU, matching jax.nn.gelu(approximate=False)
  return 0.5f * x * (1.0f + erff(x * 0.70710678118654752440f));
}

// -----------------------------------------------------------------------------
// WMMA f32 GEMM:  out[N,KOUT] = act(A[N,DIM] @ W[DIM,KOUT] + bias)
// One 16x16 output tile per wave; 8 waves / 256-thread block.
// V_WMMA_F32_16X16X4_F32 operand layouts per CDNA5 ISA 7.12.2.
// -----------------------------------------------------------------------------
template <int DIM, int KOUT, bool GELU>
__global__ __launch_bounds__(256) void gemm_bias_act(
    const float* __restrict__ A, const float* __restrict__ W,
    const float* __restrict__ bias, float* __restrict__ out, int nrows) {
  constexpr int CT  = KOUT / 16;  // column tiles per row tile
  constexpr int RPB = 8 / CT;     // row tiles per block (8 waves)

  const int lane    = threadIdx.x & 31;
  const int wave    = threadIdx.x >> 5;
  const int colTile = wave % CT;
  const int rowTile = blockIdx.x * RPB + wave / CT;
  const int row0    = rowTile * 16;
  if (row0 >= nrows) return;  // wave-uniform: EXEC stays all-ones for WMMA

  const int m  = lane & 15;  // A row within tile / B,C column within tile
  const int hi = lane >> 4;  // selects K pair (A/B) and M half (C/D)

  const float* __restrict__ arow = A + (size_t)(row0 + m) * DIM;
  const int col = colTile * 16 + m;

  v8f c = {};
#pragma unroll
  for (int k0 = 0; k0 < DIM; k0 += 4) {
    const int kk = k0 + 2 * hi;
    v2f av = *(const v2f*)(arow + kk);  // 8B aligned (DIM, kk even)
    v2f bv;
    bv.x = W[(size_t)(kk + 0) * KOUT + col];
    bv.y = W[(size_t)(kk + 1) * KOUT + col];
    c = __builtin_amdgcn_wmma_f32_16x16x4_f32(
        /*neg_a=*/false, av, /*neg_b=*/false, bv,
        /*c_mod=*/(short)0, c, /*reuse_a=*/false, /*reuse_b=*/false);
  }

  const float b = bias[col];
#pragma unroll
  for (int i = 0; i < 8; ++i) {
    const int row = row0 + i + hi * 8;
    float v = c[i] + b;
    if (GELU) v = gelu_erf(v);
    out[(size_t)row * KOUT + col] = v;
  }
}

// -----------------------------------------------------------------------------
// Edge scores: wave-per-edge. es[e] = exp( sum_d leaky_relu(q[dst]+p[src])*a_d )
// (unstabilized exp, matching the reference). No atomics here.
// -----------------------------------------------------------------------------
template <int DIM>
__global__ __launch_bounds__(256) void edge_scores(
    const float* __restrict__ q, const float* __restrict__ p,
    const float* __restrict__ a, const int* __restrict__ src,
    const int* __restrict__ dst, float* __restrict__ es, int nE) {
  constexpr int PER = DIM / 32;
  typedef typename VecT<PER>::T VT;
  const int lane = threadIdx.x & 31;
  const int e = blockIdx.x * (blockDim.x >> 5) + (threadIdx.x >> 5);
  if (e >= nE) return;
  const int s = src[e];
  const int d = dst[e];
  const int c0 = lane * PER;
  VT qv = *(const VT*)(q + (size_t)d * DIM + c0);
  VT pv = *(const VT*)(p + (size_t)s * DIM + c0);
  VT av = *(const VT*)(a + c0);
  float sum = 0.f;
#pragma unroll
  for (int i = 0; i < PER; ++i) {
    float t = qv[i] + pv[i];
    t = (t > 0.f) ? t : LEAKY_SLOPE * t;
    sum += t * av[i];
  }
#pragma unroll
  for (int off = 16; off > 0; off >>= 1) sum += __shfl_xor(sum, off, 32);
  if (lane == 0) es[e] = expf(sum);
}

// -----------------------------------------------------------------------------
// CSR construction (built once; both GAT layers share the graph).
// -----------------------------------------------------------------------------
__global__ __launch_bounds__(256) void zero_i32(int* __restrict__ b, int n) {
  const int i = blockIdx.x * blockDim.x + threadIdx.x;
  if (i < n) b[i] = 0;
}

__global__ __launch_bounds__(256) void dst_histogram(
    const int* __restrict__ dst, int* __restrict__ counts, int nE) {
  const int e = blockIdx.x * blockDim.x + threadIdx.x;
  if (e < nE) atomicAdd(&counts[dst[e]], 1);
}

// Single-block hierarchical exclusive scan: shfl wave-scan + LDS wave-sum combine.
__global__ __launch_bounds__(1024) void exclusive_scan_block(
    const int* __restrict__ counts, int* __restrict__ rowptr,
    int* __restrict__ cursor, int n) {
  __shared__ int wsum[32];
  const int lane = threadIdx.x & 31;
  const int w    = threadIdx.x >> 5;
  int carry = 0;
  for (int base = 0; base < n; base += 1024) {
    const int i = base + threadIdx.x;
    const int v = (i < n) ? counts[i] : 0;
    int incl = v;
#pragma unroll
    for (int off = 1; off < 32; off <<= 1) {
      int t = __shfl_up(incl, off, 32);
      if (lane >= off) incl += t;
    }
    if (lane == 31) wsum[w] = incl;
    __syncthreads();
    if (w == 0) {
      int ts = wsum[lane];
#pragma unroll
      for (int off = 1; off < 32; off <<= 1) {
        int u = __shfl_up(ts, off, 32);
        if (lane >= off) ts += u;
      }
      wsum[lane] = ts;
    }
    __syncthreads();
    const int waveoff = (w > 0) ? wsum[w - 1] : 0;
    const int total   = wsum[31];
    const int excl    = carry + waveoff + incl - v;
    if (i < n) {
      rowptr[i] = excl;
      cursor[i] = excl;
    }
    carry += total;
    __syncthreads();  // protect wsum before next chunk overwrites it
  }
  if (threadIdx.x == 0) rowptr[n] = carry;
}

__global__ __launch_bounds__(256) void csr_scatter(
    const int* __restrict__ src, const int* __restrict__ dst,
    int* __restrict__ cursor, int* __restrict__ eid, int* __restrict__ esrc,
    int nE) {
  const int e = blockIdx.x * blockDim.x + threadIdx.x;
  if (e < nE) {
    const int d   = dst[e];
    const int pos = atomicAdd(&cursor[d], 1);
    eid[pos]  = e;
    esrc[pos] = src[e];
  }
}

// -----------------------------------------------------------------------------
// Aggregation, wave-per-destination-node, no float atomics:
//   out[d] = ( sum_{e in in(d)} es_e * p[src_e] ) / ( sum es_e )
// which equals sum alpha_e * p[src_e] from the reference.
// -----------------------------------------------------------------------------
template <int DIM>
__global__ __launch_bounds__(256) void gat_aggregate_csr(
    const float* __restrict__ p, const float* __restrict__ es,
    const int* __restrict__ rowptr, const int* __restrict__ eid,
    const int* __restrict__ esrc, float* __restrict__ out, int n) {
  constexpr int PER = DIM / 32;
  typedef typename VecT<PER>::T VT;
  const int lane = threadIdx.x & 31;
  const int d = blockIdx.x * (blockDim.x >> 5) + (threadIdx.x >> 5);
  if (d >= n) return;  // wave-uniform
  const int beg = rowptr[d];
  const int end = rowptr[d + 1];
  const int c0 = lane * PER;

  float acc[PER];
#pragma unroll
  for (int i = 0; i < PER; ++i) acc[i] = 0.f;
  float ssum = 0.f;

  for (int idx = beg; idx < end; ++idx) {
    const int e = eid[idx];
    const int s = esrc[idx];
    if (idx + 1 < end) {  // prefetch next gathered row (global_prefetch_b8)
      __builtin_prefetch(p + (size_t)esrc[idx + 1] * DIM + c0, 0, 1);
    }
    const float w = es[e];
    VT pv = *(const VT*)(p + (size_t)s * DIM + c0);
#pragma unroll
    for (int i = 0; i < PER; ++i) acc[i] += w * pv[i];
    ssum += w;
  }
  const float inv = (ssum != 0.f) ? 1.f / ssum : 0.f;  // empty bucket -> 0
  float* __restrict__ orow = out + (size_t)d * DIM + c0;
#pragma unroll
  for (int i = 0; i < PER; ++i) orow[i] = acc[i] * inv;
}

// -----------------------------------------------------------------------------
template <int DIM, bool GELU>
__global__ __launch_bounds__(256) void bias_act(
    const float* __restrict__ in, const float* __restrict__ bias,
    float* __restrict__ out, int n) {
  const int i = blockIdx.x * blockDim.x + threadIdx.x;
  if (i < n) {
    float v = in[i] + bias[i & (DIM - 1)];
    if (GELU) v = gelu_erf(v);
    out[i] = v;
  }
}

// -----------------------------------------------------------------------------
extern "C" void kernel_launch(void* const* d_in, const int* in_sizes, int n_in,
                              void* d_out, int out_size, void* d_ws,
                              size_t ws_size, hipStream_t stream) {
  const float* x    = (const float*)d_in[0];
  const float* W0   = (const float*)d_in[1];
  const float* b0   = (const float*)d_in[2];
  const float* Wq1  = (const float*)d_in[3];
  const float* bq1  = (const float*)d_in[4];
  const float* Wp1  = (const float*)d_in[5];
  const float* bp1  = (const float*)d_in[6];
  const float* a1   = (const float*)d_in[7];
  const float* bg2  = (const float*)d_in[8];
  const float* Wq2  = (const float*)d_in[9];
  const float* bq2  = (const float*)d_in[10];
  const float* Wp2  = (const float*)d_in[11];
  const float* bp2  = (const float*)d_in[12];
  const float* a2   = (const float*)d_in[13];
  const float* bout = (const float*)d_in[14];
  const int*   src  = (const int*)d_in[15];
  const int*   dst  = (const int*)d_in[16];

  const int N = in_sizes[0] / 128;  // 50000
  const int E = in_sizes[15];       // 800000

  // Workspace layout; float region then int region.
  float* m1  = (float*)d_ws;            // [N,128] m1, later m2
  float* q1  = m1 + (size_t)N * 128;    // [N,128] q1, later q2 [N,64]
  float* p1  = q1 + (size_t)N * 128;    // [N,128] p1, later p2 [N,64]
  float* h1  = p1 + (size_t)N * 128;    // [N,128] h1, later h2 [N,64]
  float* es  = h1 + (size_t)N * 128;    // [E]
  int* counts = (int*)(es + (size_t)E); // [N]
  int* rowptr = counts + N;             // [N+1]
  int* cursor = rowptr + (N + 1);       // [N]
  int* eid    = cursor + N;             // [E]
  int* esrc   = eid + E;                // [E]

  const int rowTiles   = (N + 15) / 16;  // 3125
  const int edgeBlocks = (E + 7) / 8;    // wave-per-edge, 8 waves/block
  const int nodeBlocks = (N + 7) / 8;    // wave-per-node, 8 waves/block
  const int nh = N * 128;
  const int no = N * 64;

  // ---- CSR build (shared by both GAT layers)
  zero_i32<<<(N + 255) / 256, 256, 0, stream>>>(counts, N);
  dst_histogram<<<(E + 255) / 256, 256, 0, stream>>>(dst, counts, E);
  exclusive_scan_block<<<1, 1024, 0, stream>>>(counts, rowptr, cursor, N);
  csr_scatter<<<(E + 255) / 256, 256, 0, stream>>>(src, dst, cursor, eid, esrc, E);

  // ---- layer 0: m1 = gelu(x @ W0 + b0)
  gemm_bias_act<128, 128, true><<<rowTiles, 256, 0, stream>>>(x, W0, b0, m1, N);

  // ---- GAT block 1
  gemm_bias_act<128, 128, false><<<rowTiles, 256, 0, stream>>>(m1, Wq1, bq1, q1, N);
  gemm_bias_act<128, 128, false><<<rowTiles, 256, 0, stream>>>(m1, Wp1, bp1, p1, N);
  edge_scores<128><<<edgeBlocks, 256, 0, stream>>>(q1, p1, a1, src, dst, es, E);
  gat_aggregate_csr<128><<<nodeBlocks, 256, 0, stream>>>(p1, es, rowptr, eid, esrc, h1, N);

  // ---- m2 = gelu(h1 + bg2)  (into m1's buffer)
  bias_act<128, true><<<(nh + 255) / 256, 256, 0, stream>>>(h1, bg2, m1, nh);

  // ---- GAT block 2 (D_OUT = 64)
  gemm_bias_act<128, 64, false><<<(rowTiles + 1) / 2, 256, 0, stream>>>(m1, Wq2, bq2, q1, N);
  gemm_bias_act<128, 64, false><<<(rowTiles + 1) / 2, 256, 0, stream>>>(m1, Wp2, bp2, p1, N);
  edge_scores<64><<<edgeBlocks, 256, 0, stream>>>(q1, p1, a2, src, dst, es, E);
  gat_aggregate_csr<64><<<nodeBlocks, 256, 0, stream>>>(p1, es, rowptr, eid, esrc, h1, N);

  // ---- out = h2 + b_out
  bias_act<64, false><<<(no + 255) / 256, 256, 0, stream>>>(h1, bout, (float*)d_out, no);
}